// FragmentPositionDistribution1_79654463472195
// MI455X (gfx1250) — compile-verified
//
#include <hip/hip_runtime.h>
#include <hip/hip_bf16.h>

typedef float v2f __attribute__((ext_vector_type(2)));
typedef float v8f __attribute__((ext_vector_type(8)));
typedef int   v2i __attribute__((ext_vector_type(2)));

// ---------------------------------------------------------------------------
// Kernel 1: Esum[k*G + g] = sum_f exp(bincounts[g,f]*dw[k] + baseline[genes_oi[g],f] + cm[k])
// Reduction over f runs on the matrix pipe: A (16x4 f32) holds exp values for
// 16 (k,g) rows x 4 f-values, B = ones(4x16), accumulate D = A*B + C.
// Main loop is branch-free (EXEC stays all-ones by construction); the tail
// chunk uses clamped indices + multiplicative masks (selects, no divergence).
// ---------------------------------------------------------------------------
__global__ void expsum_wmma_kernel(const float* __restrict__ bincounts,  // G*F
                                   const float* __restrict__ baseline,   // N_GENES*F
                                   const float* __restrict__ dw,         // K
                                   const float* __restrict__ cm,         // K
                                   const int*   __restrict__ genes_oi,   // G
                                   float*       __restrict__ Esum,       // K*G
                                   int G, int F, int K) {
    const int flat  = blockIdx.x * blockDim.x + threadIdx.x;
    const int wave  = flat >> 5;           // uniform within a wave
    const int lane  = threadIdx.x & 31;
    const int nrows = K * G;
    const int rbase = wave * 16;
    if (rbase >= nrows) return;            // wave-uniform exit

    // A-matrix layout (32-bit, 16x4): lanes 0-15 -> M=lane, K=0,1 ; lanes 16-31 -> M=lane-16, K=2,3
    const int  r     = rbase + (lane & 15);
    const int  rc    = (r < nrows) ? r : (nrows - 1);    // clamp: select, no branch
    const int  k     = rc / G;
    const int  g     = rc - k * G;
    const float dwv  = dw[k];
    const float cmv  = cm[k];
    const float* bcrow = bincounts + (size_t)g * F;
    const float* blrow = baseline  + (size_t)genes_oi[g] * F;
    const int fhalf  = (lane >> 4) * 2;    // 0 for lanes 0-15, 2 for lanes 16-31

    v8f acc = {0.f, 0.f, 0.f, 0.f, 0.f, 0.f, 0.f, 0.f};
    v2f ones; ones.x = 1.0f; ones.y = 1.0f;

    const int fmain = F & ~3;              // multiple of 4: guard-free main loop
    for (int f0 = 0; f0 < fmain; f0 += 4) {
        const int f = f0 + fhalf;
        v2f a;
        a.x = __expf(bcrow[f]     * dwv + blrow[f]     + cmv);
        a.y = __expf(bcrow[f + 1] * dwv + blrow[f + 1] + cmv);
        // D(16x16) = A(16x4) * ones(4x16) + C : every column = row sums of A
        acc = __builtin_amdgcn_wmma_f32_16x16x4_f32(
                  false, a, false, ones, (short)0, acc, false, false);
    }
    if (fmain < F) {                       // tail (F % 4 != 0): masked, branch-free
        const int f   = fmain + fhalf;
        const int f0c = (f     < F) ? f     : F - 1;
        const int f1c = (f + 1 < F) ? f + 1 : F - 1;
        const float m0 = (f     < F) ? 1.0f : 0.0f;
        const float m1 = (f + 1 < F) ? 1.0f : 0.0f;
        v2f a;
        a.x = m0 * __expf(bcrow[f0c] * dwv + blrow[f0c] + cmv);
        a.y = m1 * __expf(bcrow[f1c] * dwv + blrow[f1c] + cmv);
        acc = __builtin_amdgcn_wmma_f32_16x16x4_f32(
                  false, a, false, ones, (short)0, acc, false, false);
    }

    // C/D layout: lane 0 column N=0 holds M=0..7 in VGPR 0..7; lane 16 holds M=8..15.
    if (lane == 0 || lane == 16) {
        const int rowbase = rbase + (lane >> 4) * 8;
        #pragma unroll
        for (int m = 0; m < 8; ++m) {
            const int rr = rowbase + m;
            if (rr < nrows) Esum[rr] = acc[m];
        }
    }
}

// ---------------------------------------------------------------------------
// Kernel 2: out[c*G + g] = -Esum[labels[c]*G + g]
// ---------------------------------------------------------------------------
__global__ void init_out_kernel(const int* __restrict__ labels,
                                const float* __restrict__ Esum,
                                float* __restrict__ out, int C, int G) {
    const int i = blockIdx.x * blockDim.x + threadIdx.x;
    if (i >= C * G) return;
    const int c = i / G;
    const int g = i - c * G;
    out[i] = -Esum[labels[c] * G + g];
}

// ---------------------------------------------------------------------------
// Kernel 3: per-fragment pass, 2 fragments per thread (b64 index loads).
//   old = atomicAdd(count[bin], 1)      (u16 packed 2-per-u32: fits global L2)
//   out[c,g] += u(c,g,f) - ln(old + 1)  (sum over fragments of ln(old+1) == ln(count!))
// ---------------------------------------------------------------------------
__global__ void frag_kernel(const int* __restrict__ cgix,       // NFRAG
                            const int* __restrict__ binix,      // NFRAG
                            const int* __restrict__ labels,     // C
                            const int* __restrict__ genes_oi,   // G
                            const float* __restrict__ bincounts,// G*F
                            const float* __restrict__ baseline, // N_GENES*F
                            const float* __restrict__ dw,       // K
                            const float* __restrict__ cm,       // K
                            unsigned int* __restrict__ counts,  // packed 2x u16 per u32
                            float* __restrict__ out,            // C*G
                            int G, int F, long long nfrag) {
    const long long tid    = (long long)blockIdx.x * blockDim.x + threadIdx.x;
    const long long stride = (long long)gridDim.x * blockDim.x;
    const long long npair  = (nfrag + 1) >> 1;

    for (long long p = tid; p < npair; p += stride) {
        const long long base = p << 1;
        // coalesced 8-byte index loads (pair of fragments)
        const v2i cg2 = *(const v2i*)(cgix  + base);
        const v2i bx2 = *(const v2i*)(binix + base);
        #pragma unroll
        for (int j = 0; j < 2; ++j) {
            const long long i = base + j;
            if (i >= nfrag) break;
            const int cg = (j == 0) ? cg2.x : cg2.y;
            const int f  = (j == 0) ? bx2.x : bx2.y;
            const int c  = cg / G;
            const int g  = cg - c * G;

            const size_t bin = (size_t)cg * (size_t)F + (size_t)f;
            const unsigned int sh  = (unsigned int)(bin & 1) * 16u;
            const unsigned int old = atomicAdd(&counts[bin >> 1], 1u << sh);
            const unsigned int oldcnt = (old >> sh) & 0xFFFFu;

            const int kk = labels[c];
            const float u = bincounts[(size_t)g * F + f] * dw[kk]
                          + baseline[(size_t)genes_oi[g] * F + f]
                          + cm[kk];
            const float lg = (oldcnt != 0u) ? __logf((float)(oldcnt + 1u)) : 0.0f;
            atomicAdd(&out[cg], u - lg);
        }
    }
}

extern "C" void kernel_launch(void* const* d_in, const int* in_sizes, int n_in,
                              void* d_out, int out_size, void* d_ws, size_t ws_size,
                              hipStream_t stream) {
    // Input order: bincounts, baseline_weight, differential_weight, cluster_modifier,
    //              genes_oi, labels, local_cellxgene_ix, binix
    const float* bincounts = (const float*)d_in[0];
    const float* baseline  = (const float*)d_in[1];
    const float* dw        = (const float*)d_in[2];
    const float* cm        = (const float*)d_in[3];
    const int*   genes_oi  = (const int*)d_in[4];
    const int*   labels    = (const int*)d_in[5];
    const int*   cgix      = (const int*)d_in[6];
    const int*   binix     = (const int*)d_in[7];
    float*       out       = (float*)d_out;

    const int G = in_sizes[4];
    const int F = in_sizes[0] / G;
    const int C = in_sizes[5];
    const int K = in_sizes[3];
    const long long NFRAG = in_sizes[6];

    // Workspace layout: [packed u16 counts][Esum (K*G floats)]
    const size_t nbins      = (size_t)C * (size_t)G * (size_t)F;
    const size_t countWords = (nbins + 1) / 2;
    size_t countBytes       = countWords * sizeof(unsigned int);
    countBytes              = (countBytes + 255) & ~(size_t)255;
    unsigned int* counts    = (unsigned int*)d_ws;
    float*        Esum      = (float*)((char*)d_ws + countBytes);

    // 1) zero the histogram (graph-capture-safe stream memset)
    hipMemsetAsync(counts, 0, countWords * sizeof(unsigned int), stream);

    // 2) Esum via WMMA row-sum reduction: one 16-row tile per wave
    {
        const int nrows  = K * G;
        const int nwaves = (nrows + 15) / 16;
        const int nthr   = nwaves * 32;
        const int blocks = (nthr + 255) / 256;
        expsum_wmma_kernel<<<blocks, 256, 0, stream>>>(
            bincounts, baseline, dw, cm, genes_oi, Esum, G, F, K);
    }

    // 3) out[c,g] = -Esum[labels[c], g]
    {
        const int n = C * G;
        init_out_kernel<<<(n + 255) / 256, 256, 0, stream>>>(labels, Esum, out, C, G);
    }

    // 4) fragment scatter pass (2 fragments per thread)
    {
        const long long npair  = (NFRAG + 1) >> 1;
        long long blocks = (npair + 255) / 256;
        if (blocks > 65535LL * 16) blocks = 65535LL * 16;  // plenty; grid-stride covers rest
        frag_kernel<<<(int)blocks, 256, 0, stream>>>(
            cgix, binix, labels, genes_oi, bincounts, baseline, dw, cm,
            counts, out, G, F, NFRAG);
    }
}